// GFCNB_17867063951702
// MI455X (gfx1250) — compile-verified
//
#include <hip/hip_runtime.h>
#include <hip/hip_bf16.h>
#include <math.h>

// ---------------------------------------------------------------------------
// GFCNB (SplineConv graph U-net) for MI455X / gfx1250.
//
// Roofline: ~16.6 GFLOP fp32 of dense GEMM (negligible on matrix cores) vs
// ~1.5-2 GB of trans materialization + edge gather/scatter traffic; with
// 23.3 TB/s HBM and a 192 MB L2 (largest trans block = 210 MB, edge gathers
// mostly L2-resident) the net is scatter/bandwidth bound.  Therefore:
//  * dense einsum  trans = x @ W[kk]  and  x @ root  use V_WMMA_F32_16X16X4_F32
//    (fp32 matrix cores -> exact reference numerics, GEMM time ~0)
//  * edge pass: 1 thread/edge, 4 bilinear gathers of Cout-contiguous rows,
//    global_atomic_add_f32 scatter (L2 atomics)
//  * BN reduce via LDS ds_add_f32; segment-max via monotone-uint atomicMax.
// ---------------------------------------------------------------------------

typedef float v2f __attribute__((ext_vector_type(2)));
typedef float v8f __attribute__((ext_vector_type(8)));

#define EPS_BN 1e-5f

static constexpr int cN0 = 65536, cN1 = 32768, cN2 = 16384, cN3 = 8192;
static constexpr int cE0 = 524288, cE1 = 262144, cE2 = 131072, cE3 = 65536;

// ------------------------------ device helpers -----------------------------

__device__ __forceinline__ unsigned enc_f32(float f) {
  unsigned u = __float_as_uint(f);
  return (u >> 31) ? ~u : (u | 0x80000000u);   // monotone mapping for max
}
__device__ __forceinline__ float dec_f32(unsigned e) {
  unsigned u = (e >> 31) ? (e & 0x7FFFFFFFu) : ~e;
  return __uint_as_float(u);
}

// ------------------------------ fill kernels -------------------------------

__global__ void k_fill_f32(float* p, float v, size_t n) {
  size_t i = (size_t)blockIdx.x * blockDim.x + threadIdx.x;
  if (i < n) p[i] = v;
}
__global__ void k_fill_u32(unsigned* p, unsigned v, size_t n) {
  size_t i = (size_t)blockIdx.x * blockDim.x + threadIdx.x;
  if (i < n) p[i] = v;
}

// ------------------------------ cartesian ----------------------------------

__global__ void k_cart_absmax(const int* __restrict__ ei, int E,
                              const float* __restrict__ pos, unsigned* mx) {
  int e = blockIdx.x * blockDim.x + threadIdx.x;
  if (e >= E) return;
  int s = ei[e], d = ei[E + e];
  float cx = pos[2 * d] - pos[2 * s];
  float cy = pos[2 * d + 1] - pos[2 * s + 1];
  float a = fmaxf(fabsf(cx), fabsf(cy));      // a >= 0 -> uint order == float order
  atomicMax(mx, __float_as_uint(a));
}

__global__ void k_cart_norm(const int* __restrict__ ei, int E,
                            const float* __restrict__ pos,
                            const unsigned* __restrict__ mx,
                            float* __restrict__ ea) {
  int e = blockIdx.x * blockDim.x + threadIdx.x;
  if (e >= E) return;
  float m = __uint_as_float(*mx);
  float inv = 1.0f / (2.0f * m);
  int s = ei[e], d = ei[E + e];
  ea[2 * e]     = (pos[2 * d]     - pos[2 * s])     * inv + 0.5f;
  ea[2 * e + 1] = (pos[2 * d + 1] - pos[2 * s + 1]) * inv + 0.5f;
}

// ------------------------------ degree -------------------------------------

__global__ void k_deg(const int* __restrict__ ei, int E, float* __restrict__ deg) {
  int e = blockIdx.x * blockDim.x + threadIdx.x;
  if (e >= E) return;
  atomicAdd(&deg[ei[E + e]], 1.0f);
}

// -------------------- WMMA fp32 GEMM: C[M,Nc] = A[M,Kc]@B[Kc,Nc] -----------
// One wave per 16x16 output tile; V_WMMA_F32_16X16X4_F32, K stepped by 4.
// Layouts per CDNA5 ISA 7.12.2:
//   A 16x4 : lanes 0-15 -> K=0,1 in the 2 VGPRs; lanes 16-31 -> K=2,3
//   B 4x16 : rows striped across lanes (K=0/2 in VGPR0, K=1/3 in VGPR1)
//   D 16x16: VGPR v -> row v (lanes 0-15) / row v+8 (lanes 16-31)

__global__ void k_wmma_gemm(const float* __restrict__ A, int lda,
                            const float* __restrict__ B, int ldb,
                            float* __restrict__ C, int ldc,
                            int M, int Nc, int Kc) {
  int gtid = blockIdx.x * blockDim.x + threadIdx.x;
  int wave = gtid >> 5;              // wave-uniform
  int lane = gtid & 31;
  int tilesN = Nc >> 4;
  int ntiles = (M >> 4) * tilesN;
  if (wave >= ntiles) return;        // uniform branch -> EXEC all-ones in WMMA
  int tm = wave / tilesN, tn = wave % tilesN;
  int half = lane >> 4;              // 0: K=0,1  1: K=2,3
  int l = lane & 15;
  const float* Arow = A + (size_t)(tm * 16 + l) * lda;
  const float* Bcol = B + tn * 16 + l;
  v8f acc = {0.f, 0.f, 0.f, 0.f, 0.f, 0.f, 0.f, 0.f};
  for (int k0 = 0; k0 < Kc; k0 += 4) {
    int ka = k0 + 2 * half;
    v2f a = { Arow[ka], Arow[ka + 1] };
    v2f b = { Bcol[(size_t)ka * ldb], Bcol[(size_t)(ka + 1) * ldb] };
    acc = __builtin_amdgcn_wmma_f32_16x16x4_f32(
        /*neg_a=*/false, a, /*neg_b=*/false, b,
        /*c_mod=*/(short)0, acc, /*reuse_a=*/false, /*reuse_b=*/false);
  }
  float* Crow = C + (size_t)(tm * 16 + 8 * half) * ldc + tn * 16 + l;
#pragma unroll
  for (int v = 0; v < 8; ++v) Crow[(size_t)v * ldc] = acc[v];
}

// ----------------- Cin==1 "GEMM": C[m, j] = x[m] * w[j] --------------------

__global__ void k_cin1_outer(const float* __restrict__ x, const float* __restrict__ w,
                             float* __restrict__ C, int M, int Ntot) {
  size_t i = (size_t)blockIdx.x * blockDim.x + threadIdx.x;
  if (i >= (size_t)M * Ntot) return;
  int m = (int)(i / Ntot), j = (int)(i % Ntot);
  C[i] = x[m] * w[j];
}

// ------------- small-N GEMM: out[m*ldo+kk] = dot(x[m,:Cin], W[kk,:Cin]) ----

__global__ void k_dot_small(const float* __restrict__ x, int Cin,
                            const float* __restrict__ W, int K2,
                            float* __restrict__ out, int ldo, int M) {
  size_t i = (size_t)blockIdx.x * blockDim.x + threadIdx.x;
  if (i >= (size_t)M * K2) return;
  int m = (int)(i / K2), kk = (int)(i % K2);
  const float* xr = x + (size_t)m * Cin;
  const float* wr = W + (size_t)kk * Cin;
  float s = 0.f;
  for (int c = 0; c < Cin; ++c) s += xr[c] * wr[c];
  out[(size_t)m * ldo + kk] = s;
}

// ------------------------------ edge message pass --------------------------

__global__ void k_edge_msg(const int* __restrict__ ei, int E,
                           const float* __restrict__ ea,
                           const float* __restrict__ trans,
                           int K, int Cout,
                           float* __restrict__ agg) {
  int e = blockIdx.x * blockDim.x + threadIdx.x;
  if (e >= E) return;
  int src = ei[e], dst = ei[E + e];
  float Km1 = (float)(K - 1);
  float fr0 = ea[2 * e] * Km1, fr1 = ea[2 * e + 1] * Km1;
  float bo0 = fminf(fmaxf(floorf(fr0), 0.f), Km1);
  float bo1 = fminf(fmaxf(floorf(fr1), 0.f), Km1);
  float f0 = fr0 - bo0, f1 = fr1 - bo1;
  int b0 = (int)bo0, b1 = (int)bo1;
  int t0 = min(b0 + 1, K - 1), t1 = min(b1 + 1, K - 1);
  float w00 = (1.f - f0) * (1.f - f1), w10 = f0 * (1.f - f1);
  float w01 = (1.f - f0) * f1,         w11 = f0 * f1;
  const float* tr = trans + (size_t)src * (K * K) * Cout;
  const float* p00 = tr + (size_t)(b0 + b1 * K) * Cout;
  const float* p10 = tr + (size_t)(t0 + b1 * K) * Cout;
  const float* p01 = tr + (size_t)(b0 + t1 * K) * Cout;
  const float* p11 = tr + (size_t)(t0 + t1 * K) * Cout;
  float* o = agg + (size_t)dst * Cout;
  for (int c = 0; c < Cout; ++c) {
    float m = w00 * p00[c] + w10 * p10[c] + w01 * p01[c] + w11 * p11[c];
    atomicAdd(&o[c], m);
  }
}

// ------------------- combine: act(agg/deg + x@root + bias) -----------------

__global__ void k_combine(const float* __restrict__ agg, const float* __restrict__ xr,
                          const float* __restrict__ deg, const float* __restrict__ bias,
                          float* __restrict__ out, int N, int C, int elu) {
  size_t i = (size_t)blockIdx.x * blockDim.x + threadIdx.x;
  if (i >= (size_t)N * C) return;
  int n = (int)(i / C), c = (int)(i % C);
  float v = agg[i] / fmaxf(deg[n], 1.0f) + xr[i] + bias[c];
  if (elu) v = (v > 0.f) ? v : (expf(v) - 1.0f);
  out[i] = v;
}

// ------------------------------ batch norm ---------------------------------

__global__ void k_bn_reduce(const float* __restrict__ x, int N, int C,
                            float* __restrict__ gsum, float* __restrict__ gssq) {
  __shared__ float s_sum[128];
  __shared__ float s_ssq[128];
  for (int c = threadIdx.x; c < C; c += blockDim.x) { s_sum[c] = 0.f; s_ssq[c] = 0.f; }
  __syncthreads();
  size_t total = (size_t)N * C;
  size_t stride = (size_t)gridDim.x * blockDim.x;
  for (size_t i = (size_t)blockIdx.x * blockDim.x + threadIdx.x; i < total; i += stride) {
    int c = (int)(i % C);
    float v = x[i];
    atomicAdd(&s_sum[c], v);        // ds_add_f32
    atomicAdd(&s_ssq[c], v * v);
  }
  __syncthreads();
  for (int c = threadIdx.x; c < C; c += blockDim.x) {
    atomicAdd(&gsum[c], s_sum[c]);
    atomicAdd(&gssq[c], s_ssq[c]);
  }
}

__global__ void k_bn_apply(float* __restrict__ x, int N, int C,
                           const float* __restrict__ gsum, const float* __restrict__ gssq,
                           const float* __restrict__ g, const float* __restrict__ b) {
  size_t i = (size_t)blockIdx.x * blockDim.x + threadIdx.x;
  if (i >= (size_t)N * C) return;
  int c = (int)(i % C);
  float invN = 1.0f / (float)N;
  float mu = gsum[c] * invN;
  float var = gssq[c] * invN - mu * mu;
  x[i] = g[c] * (x[i] - mu) * rsqrtf(var + EPS_BN) + b[c];
}

// ------------------------------ segment max pool ---------------------------

__global__ void k_pool_max(const float* __restrict__ x, const int* __restrict__ cl,
                           int N, int C, unsigned* __restrict__ pooled) {
  int n = blockIdx.x * blockDim.x + threadIdx.x;
  if (n >= N) return;
  int c = cl[n];
  const float* xr = x + (size_t)n * C;
  unsigned* o = pooled + (size_t)c * C;
  for (int ch = 0; ch < C; ++ch) atomicMax(&o[ch], enc_f32(xr[ch]));
}

__global__ void k_pool_decode(unsigned* __restrict__ p, size_t n) {
  size_t i = (size_t)blockIdx.x * blockDim.x + threadIdx.x;
  if (i >= n) return;
  unsigned e = p[i];
  ((float*)p)[i] = dec_f32(e);
}

// ------------------------------ segment mean of pos ------------------------

__global__ void k_seg_pos(const float* __restrict__ pos, const int* __restrict__ cl,
                          int N, float* __restrict__ psum, float* __restrict__ cnt) {
  int n = blockIdx.x * blockDim.x + threadIdx.x;
  if (n >= N) return;
  int c = cl[n];
  atomicAdd(&psum[2 * c],     pos[2 * n]);
  atomicAdd(&psum[2 * c + 1], pos[2 * n + 1]);
  atomicAdd(&cnt[c], 1.0f);
}

__global__ void k_seg_pos_div(float* __restrict__ psum, const float* __restrict__ cnt, int Nc) {
  int i = blockIdx.x * blockDim.x + threadIdx.x;
  if (i >= Nc) return;
  float inv = 1.0f / fmaxf(cnt[i], 1.0f);
  psum[2 * i] *= inv;
  psum[2 * i + 1] *= inv;
}

// ------------------------------ recover grid -------------------------------

__global__ void k_recover(const float* __restrict__ xs, const int* __restrict__ c1,
                          const int* __restrict__ c2, const int* __restrict__ c3,
                          float* __restrict__ out, int N0n, int C) {
  int n = blockIdx.x * blockDim.x + threadIdx.x;
  if (n >= N0n) return;
  int i3 = c3[c2[c1[n]]];
  const float* s = xs + (size_t)i3 * C;
  float* o = out + (size_t)n * C;
  for (int c = 0; c < C; ++c) o[c] = s[c];
}

// =========================== host orchestration ============================

static inline unsigned gdiv(size_t n, unsigned b) { return (unsigned)((n + b - 1) / b); }

static void fill_f32(float* p, float v, size_t n, hipStream_t s) {
  k_fill_f32<<<gdiv(n, 256), 256, 0, s>>>(p, v, n);
}
static void fill_u32(unsigned* p, unsigned v, size_t n, hipStream_t s) {
  k_fill_u32<<<gdiv(n, 256), 256, 0, s>>>(p, v, n);
}

static void gemm(const float* A, int lda, const float* B, int ldb,
                 float* C, int ldc, int M, int Nc, int Kc, hipStream_t s) {
  size_t tiles = (size_t)(M / 16) * (Nc / 16);
  k_wmma_gemm<<<gdiv(tiles * 32, 256), 256, 0, s>>>(A, lda, B, ldb, C, ldc, M, Nc, Kc);
}

// trans[n, kk*Cout + co] = sum_ci x[n,ci] * W[kk,ci,co]   (K2 GEMM calls)
static void trans_wmma(const float* x, const float* W, float* TRANS,
                       int N, int Cin, int Cout, int K2, hipStream_t s) {
  int ldc = K2 * Cout;
  for (int kk = 0; kk < K2; ++kk)
    gemm(x, Cin, W + (size_t)kk * Cin * Cout, Cout,
         TRANS + (size_t)kk * Cout, ldc, N, Cout, Cin, s);
}

// full WMMA spline-conv level (Cin>=16 && Cout>=16): result written to xalt
static void conv_wmma(const float* xcur, float* xalt, float* XR, float* TRANS,
                      int N, int E, const int* ei, const float* ea, const float* deg,
                      int Cin, int Cout, int K,
                      const float* W, const float* root, const float* bias,
                      int elu, hipStream_t s) {
  trans_wmma(xcur, W, TRANS, N, Cin, Cout, K * K, s);
  fill_f32(xalt, 0.f, (size_t)N * Cout, s);
  k_edge_msg<<<gdiv(E, 256), 256, 0, s>>>(ei, E, ea, TRANS, K, Cout, xalt);
  gemm(xcur, Cin, root, Cout, XR, Cout, N, Cout, Cin, s);
  k_combine<<<gdiv((size_t)N * Cout, 256), 256, 0, s>>>(xalt, XR, deg, bias, xalt, N, Cout, elu);
}

static void do_cart(const int* ei, int E, const float* pos, unsigned* mx,
                    float* ea, hipStream_t s) {
  fill_u32(mx, 0u, 1, s);
  k_cart_absmax<<<gdiv(E, 256), 256, 0, s>>>(ei, E, pos, mx);
  k_cart_norm<<<gdiv(E, 256), 256, 0, s>>>(ei, E, pos, mx, ea);
}

static void do_deg(const int* ei, int E, float* deg, int N, hipStream_t s) {
  fill_f32(deg, 0.f, (size_t)N, s);
  k_deg<<<gdiv(E, 256), 256, 0, s>>>(ei, E, deg);
}

static void do_bn(float* x, int N, int C, const float* g, const float* b,
                  float* SUM, float* SSQ, hipStream_t s) {
  fill_f32(SUM, 0.f, C, s);
  fill_f32(SSQ, 0.f, C, s);
  k_bn_reduce<<<256, 256, 0, s>>>(x, N, C, SUM, SSQ);
  k_bn_apply<<<gdiv((size_t)N * C, 256), 256, 0, s>>>(x, N, C, SUM, SSQ, g, b);
}

static void do_pool(const float* x, const int* cl, int Nprev, int Nnext, int C,
                    float* out, hipStream_t s) {
  fill_u32((unsigned*)out, 0u, (size_t)Nnext * C, s);  // 0 < enc(any finite)
  k_pool_max<<<gdiv(Nprev, 256), 256, 0, s>>>(x, cl, Nprev, C, (unsigned*)out);
  k_pool_decode<<<gdiv((size_t)Nnext * C, 256), 256, 0, s>>>((unsigned*)out, (size_t)Nnext * C);
}

static void do_segpos(const float* pos, const int* cl, int Nprev, int Nnext,
                      float* posn, float* cnt, hipStream_t s) {
  fill_f32(posn, 0.f, (size_t)2 * Nnext, s);
  fill_f32(cnt, 0.f, (size_t)Nnext, s);
  k_seg_pos<<<gdiv(Nprev, 256), 256, 0, s>>>(pos, cl, Nprev, posn, cnt);
  k_seg_pos_div<<<gdiv(Nnext, 256), 256, 0, s>>>(posn, cnt, Nnext);
}

extern "C" void kernel_launch(void* const* d_in, const int* in_sizes, int n_in,
                              void* d_out, int out_size, void* d_ws, size_t ws_size,
                              hipStream_t stream) {
  hipStream_t s = stream;
  // ---- inputs (setup_inputs insertion order; params leaves in dict order) --
  const float* x_in = (const float*)d_in[0];          // [N0,1]
  const float* pos0 = (const float*)d_in[1];          // [N0,2]
  const float *W1a = (const float*)d_in[2],  *r1a = (const float*)d_in[3],  *b1a = (const float*)d_in[4];
  const float *W1b = (const float*)d_in[5],  *r1b = (const float*)d_in[6],  *b1b = (const float*)d_in[7];
  const float *g1  = (const float*)d_in[8],  *be1 = (const float*)d_in[9];
  const float *W2a = (const float*)d_in[10], *r2a = (const float*)d_in[11], *b2a = (const float*)d_in[12];
  const float *W2b = (const float*)d_in[13], *r2b = (const float*)d_in[14], *b2b = (const float*)d_in[15];
  const float *g2  = (const float*)d_in[16], *be2 = (const float*)d_in[17];
  const float *W3a = (const float*)d_in[18], *r3a = (const float*)d_in[19], *b3a = (const float*)d_in[20];
  const float *W3b = (const float*)d_in[21], *r3b = (const float*)d_in[22], *b3b = (const float*)d_in[23];
  const float *g3  = (const float*)d_in[24], *be3 = (const float*)d_in[25];
  const float *Wsf = (const float*)d_in[26], *rsf = (const float*)d_in[27], *bsf = (const float*)d_in[28];
  const float *Wco = (const float*)d_in[29], *rco = (const float*)d_in[30], *bco = (const float*)d_in[31];
  const int* ei0 = (const int*)d_in[32];
  const int* cl1 = (const int*)d_in[33];
  const int* ei1 = (const int*)d_in[34];
  const int* cl2 = (const int*)d_in[35];
  const int* ei2 = (const int*)d_in[36];
  const int* cl3 = (const int*)d_in[37];
  const int* ei3 = (const int*)d_in[38];
  float* out = (float*)d_out;

  // ---- workspace layout (floats) ----
  float* base = (float*)d_ws;
  size_t o = 0;
  auto alloc = [&](size_t n) { float* p = base + o; o += n; return p; };
  float* TRANS = alloc((size_t)cN0 * 800);   // max trans: conv1b [N0,25,32] = 210 MB
  float* XA    = alloc((size_t)cN0 * 32);    // node-feature ping (max 2M floats)
  float* XB    = alloc((size_t)cN0 * 32);    // node-feature pong
  float* XR    = alloc((size_t)cN0 * 32);    // x@root buffer
  float* EA0   = alloc((size_t)2 * cE0);     // level-0 pseudo coords (reused at end)
  float* EAX   = alloc((size_t)2 * cE1);     // pseudo coords levels 1..3
  float* POS1  = alloc((size_t)2 * cN1);
  float* POS2  = alloc((size_t)2 * cN2);
  float* POS3  = alloc((size_t)2 * cN3);
  float* DEG   = alloc((size_t)cN0);
  float* CNT   = alloc((size_t)cN1);
  float* SUM   = alloc(128);
  float* SSQ   = alloc(128);
  unsigned* MAXU = (unsigned*)alloc(16);

  float* xcur = XA;
  float* xalt = XB;
  auto swapx = [&]() { float* t = xcur; xcur = xalt; xalt = t; };

  // ---------------- level 0 ----------------
  do_cart(ei0, cE0, pos0, MAXU, EA0, s);
  do_deg(ei0, cE0, DEG, cN0, s);

  // conv1a: Cin=1, Cout=32, K=5  (Cin==1 -> outer product, no GEMM)
  k_cin1_outer<<<gdiv((size_t)cN0 * 800, 256), 256, 0, s>>>(x_in, W1a, TRANS, cN0, 800);
  fill_f32(xalt, 0.f, (size_t)cN0 * 32, s);
  k_edge_msg<<<gdiv(cE0, 256), 256, 0, s>>>(ei0, cE0, EA0, TRANS, 5, 32, xalt);
  k_cin1_outer<<<gdiv((size_t)cN0 * 32, 256), 256, 0, s>>>(x_in, r1a, XR, cN0, 32);
  k_combine<<<gdiv((size_t)cN0 * 32, 256), 256, 0, s>>>(xalt, XR, DEG, b1a, xalt, cN0, 32, 1);
  swapx();

  // conv1b: 32->32, K=5 (WMMA)
  conv_wmma(xcur, xalt, XR, TRANS, cN0, cE0, ei0, EA0, DEG, 32, 32, 5, W1b, r1b, b1b, 1, s);
  swapx();
  do_bn(xcur, cN0, 32, g1, be1, SUM, SSQ, s);
  do_pool(xcur, cl1, cN0, cN1, 32, xalt, s);
  swapx();
  do_segpos(pos0, cl1, cN0, cN1, POS1, CNT, s);

  // ---------------- level 1 ----------------
  do_cart(ei1, cE1, POS1, MAXU, EAX, s);
  do_deg(ei1, cE1, DEG, cN1, s);
  conv_wmma(xcur, xalt, XR, TRANS, cN1, cE1, ei1, EAX, DEG, 32, 64, 3, W2a, r2a, b2a, 1, s);
  swapx();
  conv_wmma(xcur, xalt, XR, TRANS, cN1, cE1, ei1, EAX, DEG, 64, 64, 3, W2b, r2b, b2b, 1, s);
  swapx();
  do_bn(xcur, cN1, 64, g2, be2, SUM, SSQ, s);
  do_pool(xcur, cl2, cN1, cN2, 64, xalt, s);
  swapx();
  do_segpos(POS1, cl2, cN1, cN2, POS2, CNT, s);

  // ---------------- level 2 ----------------
  do_cart(ei2, cE2, POS2, MAXU, EAX, s);
  do_deg(ei2, cE2, DEG, cN2, s);
  conv_wmma(xcur, xalt, XR, TRANS, cN2, cE2, ei2, EAX, DEG, 64, 128, 3, W3a, r3a, b3a, 1, s);
  swapx();
  conv_wmma(xcur, xalt, XR, TRANS, cN2, cE2, ei2, EAX, DEG, 128, 128, 3, W3b, r3b, b3b, 1, s);
  swapx();
  do_bn(xcur, cN2, 128, g3, be3, SUM, SSQ, s);
  do_pool(xcur, cl3, cN2, cN3, 128, xalt, s);
  swapx();
  do_segpos(POS2, cl3, cN2, cN3, POS3, CNT, s);

  // ---------------- level 3: score_fr (128->32, K=5) ----------------
  do_cart(ei3, cE3, POS3, MAXU, EAX, s);
  do_deg(ei3, cE3, DEG, cN3, s);
  conv_wmma(xcur, xalt, XR, TRANS, cN3, cE3, ei3, EAX, DEG, 128, 32, 5, Wsf, rsf, bsf, 1, s);
  swapx();

  // ---------------- recover grid: x[c3][c2][c1] -> [N0,32] ----------------
  k_recover<<<gdiv(cN0, 256), 256, 0, s>>>(xcur, cl1, cl2, cl3, xalt, cN0, 32);
  swapx();

  // ---------------- convout: 32->1, K=5, no ELU ----------------
  k_dot_small<<<gdiv((size_t)cN0 * 25, 256), 256, 0, s>>>(xcur, 32, Wco, 25, TRANS, 25, cN0);
  do_deg(ei0, cE0, DEG, cN0, s);
  fill_f32(XR, 0.f, (size_t)cN0, s);
  k_edge_msg<<<gdiv(cE0, 256), 256, 0, s>>>(ei0, cE0, EA0, TRANS, 5, 1, XR);
  k_dot_small<<<gdiv((size_t)cN0, 256), 256, 0, s>>>(xcur, 32, rco, 1, xalt, 1, cN0);
  k_combine<<<gdiv((size_t)cN0, 256), 256, 0, s>>>(XR, xalt, DEG, bco, out, cN0, 1, 0);
}